// BaselineModel_87479893885203
// MI455X (gfx1250) — compile-verified
//
#include <hip/hip_runtime.h>
#include <hip/hip_bf16.h>
#include <math.h>

// Problem constants (B, S, H, T) = (64, 512, 1024, 67)
#define BB   64
#define SS   512
#define HH   1024
#define TT   67
#define TP   68      // padded trans row stride (avoid LDS bank conflicts)
#define NT   16      // WMMA N tile
#define WAVES 4      // waves per GEMM block

typedef float v2f __attribute__((ext_vector_type(2)));
typedef float v8f __attribute__((ext_vector_type(8)));

// ---------------------------------------------------------------------------
// Kernel 1: em[m, t] = leaky_relu(hidden[m, :]) @ W[:, t] + b[t]
// M = B*S = 32768 rows, N = T = 67 (5 tiles of 16, padded), K = H = 1024.
// One wave computes a 16x16 C tile with V_WMMA_F32_16X16X4_F32, K-loop of 256.
// A fragment (16x4 f32): lane l holds M = l&15, K = k0 + 2*(l>>4) .. +1  -> float2
// B fragment (4x16 f32): lane l holds N = l&15, K = k0 + 2*(l>>4) .. +1
// C tile (16x16 f32):    VGPR r, lanes 0-15: M=r, N=lane; lanes16-31: M=8+r, N=lane-16
// ---------------------------------------------------------------------------
__global__ __launch_bounds__(32 * WAVES) void emission_gemm_wmma(
    const float* __restrict__ hidden,   // [M, H]
    const float* __restrict__ W,        // [H, T]
    const float* __restrict__ bias,     // [T]
    float* __restrict__ em)             // [M, T]
{
    __shared__ float Wl[HH * NT];       // 64 KB W tile, layout [k][n]

    const int n0 = blockIdx.y * NT;

    // Stage W[:, n0:n0+16] into LDS (zero-pad columns >= 67).
    for (int i = threadIdx.x; i < HH * NT; i += blockDim.x) {
        const int k = i >> 4;
        const int n = i & 15;
        const int nn = n0 + n;
        Wl[i] = (nn < TT) ? W[k * TT + nn] : 0.0f;
    }
    __syncthreads();

    const int wave  = threadIdx.x >> 5;
    const int lane  = threadIdx.x & 31;
    const int mtile = blockIdx.x * WAVES + wave;
    const int mbase = mtile * 16;

    const int mA    = lane & 15;          // A row owned by this lane
    const int khalf = (lane >> 4) * 2;    // K sub-offset (0 or 2)
    const int nB    = lane & 15;          // B column owned by this lane

    const float* arow = hidden + (size_t)(mbase + mA) * HH;

    v8f c = {};
#pragma unroll 4
    for (int k0 = 0; k0 < HH; k0 += 4) {
        const int kk = k0 + khalf;
        // A: coalesced 8-byte load + leaky_relu(0.01)
        v2f a = *(const v2f*)(arow + kk);
        a.x = (a.x > 0.0f) ? a.x : 0.01f * a.x;
        a.y = (a.y > 0.0f) ? a.y : 0.01f * a.y;
        // B: two conflict-free ds_load_b32 from the staged W tile
        v2f bf;
        bf.x = Wl[(kk)     * NT + nB];
        bf.y = Wl[(kk + 1) * NT + nB];
        c = __builtin_amdgcn_wmma_f32_16x16x4_f32(
                /*neg_a=*/false, a, /*neg_b=*/false, bf,
                /*c_mod=*/(short)0, c, /*reuse_a=*/false, /*reuse_b=*/false);
    }

    // Epilogue: add bias, store valid columns (coalesced b32 across lanes).
    const int nglob = n0 + nB;
    if (nglob < TT) {
        const float bv = bias[nglob];
        const int mhi = (lane >> 4) * 8;
#pragma unroll
        for (int r = 0; r < 8; ++r) {
            const int mg = mbase + mhi + r;
            em[(size_t)mg * TT + nglob] = c[r] + bv;
        }
    }
}

// ---------------------------------------------------------------------------
// Kernel 2: per-batch CRF forward (logsumexp scan) fused with Viterbi forward.
// One block per batch; threads 0..66 each own one destination state t'.
// trans staged in LDS with padded stride 68. Numerator reduced in parallel.
// ---------------------------------------------------------------------------
__global__ __launch_bounds__(128) void crf_forward_fused(
    const float* __restrict__ em,          // [B, S, T]
    const int*   __restrict__ labels,      // [B, S]
    const int*   __restrict__ mask,        // [B, S]
    const float* __restrict__ start_trans, // [T]
    const float* __restrict__ end_trans,   // [T]
    const float* __restrict__ trans,       // [T, T]
    float* __restrict__ loss,              // d_out[0], pre-zeroed
    unsigned char* __restrict__ hist,      // [B, S-1, T]
    int* __restrict__ bestEnd)             // [B]
{
    __shared__ float tr[TT * TP];
    __shared__ float score[TP];
    __shared__ float vscore[TP];
    __shared__ float red[128];

    const int b   = blockIdx.x;
    const int tid = threadIdx.x;

    const float* emb = em + (size_t)b * SS * TT;
    const int* lab   = labels + b * SS;
    const int* msk   = mask + b * SS;
    unsigned char* hb = hist + (size_t)b * (SS - 1) * TT;

    for (int i = tid; i < TT * TT; i += blockDim.x) {
        const int r = i / TT;
        const int cidx = i - r * TT;
        tr[r * TP + cidx] = trans[i];
    }
    if (tid < TT) {
        const float s0 = start_trans[tid] + emb[tid];
        score[tid]  = s0;
        vscore[tid] = s0;
    }
    __syncthreads();

    // Numerator partial sums over masked steps s >= 1.
    float numpart = 0.0f;
    for (int s = 1 + tid; s < SS; s += blockDim.x) {
        if (msk[s]) {
            numpart += tr[lab[s - 1] * TP + lab[s]] + emb[(size_t)s * TT + lab[s]];
        }
    }

    // Sequential scan over time (mask is uniform per block per step).
    for (int s = 1; s < SS; ++s) {
        const int m_i = msk[s];
        if (m_i) {
            float ns = 0.0f, nv = 0.0f;
            int bidx = tid;
            if (tid < TT) {
                // Fused online logsumexp (CRF) + argmax (Viterbi) over source t.
                float mx = -INFINITY, sm = 0.0f;
                float bmax = -INFINITY;
                for (int t = 0; t < TT; ++t) {
                    const float trv = tr[t * TP + tid];
                    const float v = score[t] + trv;
                    if (v > mx) { sm = sm * __expf(mx - v) + 1.0f; mx = v; }
                    else        { sm += __expf(v - mx); }
                    const float vv = vscore[t] + trv;
                    if (vv > bmax) { bmax = vv; bidx = t; }
                }
                const float e = emb[(size_t)s * TT + tid];
                ns = mx + __logf(sm) + e;
                nv = bmax + e;
            }
            __syncthreads();
            if (tid < TT) {
                score[tid]  = ns;
                vscore[tid] = nv;
                hb[(size_t)(s - 1) * TT + tid] = (unsigned char)bidx;
            }
            __syncthreads();
        } else {
            // Masked step: state unchanged, history = identity (matches reference).
            if (tid < TT) hb[(size_t)(s - 1) * TT + tid] = (unsigned char)tid;
        }
    }

    // Reduce numerator partials.
    red[tid] = numpart;
    __syncthreads();
    for (int off = 64; off > 0; off >>= 1) {
        if (tid < off) red[tid] += red[tid + off];
        __syncthreads();
    }

    if (tid == 0) {
        int len = 0;
        for (int s = 0; s < SS; ++s) len += (msk[s] != 0);
        const int t0 = lab[0];
        const float num = red[0] + start_trans[t0] + emb[t0] + end_trans[lab[len - 1]];

        // denom = logsumexp(score + end_trans)
        float mx = -INFINITY;
        for (int t = 0; t < TT; ++t) {
            const float v = score[t] + end_trans[t];
            if (v > mx) mx = v;
        }
        float sm = 0.0f;
        for (int t = 0; t < TT; ++t) sm += __expf(score[t] + end_trans[t] - mx);
        const float denom = mx + __logf(sm);

        atomicAdd(loss, -(num - denom) * (1.0f / (float)BB));

        // Viterbi terminal argmax.
        float bm = -INFINITY; int bi = 0;
        for (int t = 0; t < TT; ++t) {
            const float v = vscore[t] + end_trans[t];
            if (v > bm) { bm = v; bi = t; }
        }
        bestEnd[b] = bi;
    }
}

// ---------------------------------------------------------------------------
// Kernel 3: Viterbi backtrack, one thread per batch (sequential gathers).
// Tags written as float (d_out is fp32).
// ---------------------------------------------------------------------------
__global__ void viterbi_backtrack(
    const unsigned char* __restrict__ hist,   // [B, S-1, T]
    const int* __restrict__ bestEnd,          // [B]
    const int* __restrict__ mask,             // [B, S]
    float* __restrict__ tags_out)             // [B, S]
{
    const int b = blockIdx.x * blockDim.x + threadIdx.x;
    if (b >= BB) return;
    const unsigned char* hb = hist + (size_t)b * (SS - 1) * TT;
    const int* msk = mask + b * SS;
    int carry = bestEnd[b];
    tags_out[b * SS + SS - 1] = msk[SS - 1] ? (float)carry : 0.0f;
    for (int s = SS - 2; s >= 0; --s) {
        carry = hb[(size_t)s * TT + carry];
        tags_out[b * SS + s] = msk[s] ? (float)carry : 0.0f;
    }
}

// ---------------------------------------------------------------------------
extern "C" void kernel_launch(void* const* d_in, const int* in_sizes, int n_in,
                              void* d_out, int out_size, void* d_ws, size_t ws_size,
                              hipStream_t stream) {
    const float* hidden      = (const float*)d_in[0];   // [B,S,H]
    const int*   labels      = (const int*)  d_in[1];   // [B,S]
    const int*   mask        = (const int*)  d_in[2];   // [B,S]
    const float* W           = (const float*)d_in[3];   // [H,T]
    const float* bias        = (const float*)d_in[4];   // [T]
    const float* start_trans = (const float*)d_in[5];   // [T]
    const float* end_trans   = (const float*)d_in[6];   // [T]
    const float* trans       = (const float*)d_in[7];   // [T,T]

    float* out  = (float*)d_out;
    float* loss = out;                       // [1]
    float* tags = out + 1;                   // [B*S]
    float* em   = out + 1 + BB * SS;         // [B*S*T]

    unsigned char* hist = (unsigned char*)d_ws;          // B*(S-1)*T bytes
    const size_t histBytes = (size_t)BB * (SS - 1) * TT; // 2,191,168 (4B aligned)
    int* bestEnd = (int*)((char*)d_ws + histBytes);

    hipMemsetAsync(loss, 0, sizeof(float), stream);

    // GEMM: M=32768 rows -> 512 blocks of 4 waves x 16 rows; 5 N-tiles.
    emission_gemm_wmma<<<dim3((BB * SS) / (16 * WAVES), (TT + NT - 1) / NT),
                         32 * WAVES, 0, stream>>>(hidden, W, bias, em);

    crf_forward_fused<<<BB, 128, 0, stream>>>(em, labels, mask, start_trans,
                                              end_trans, trans, loss, hist, bestEnd);

    viterbi_backtrack<<<1, BB, 0, stream>>>(hist, bestEnd, mask, tags);
}